// ProximityLoss_48739288875316
// MI455X (gfx1250) — compile-verified
//
#include <hip/hip_runtime.h>
#include <math.h>

// CDNA5 / gfx1250, wave32.
typedef __attribute__((ext_vector_type(2))) float v2f;
typedef __attribute__((ext_vector_type(4))) float v4f;
typedef __attribute__((ext_vector_type(8))) float v8f;

#define DDIM 1024
#define BLOCK_THREADS 256
#define WAVES_PER_BLOCK 8
#define SAMPLES_PER_BLOCK 8        // one sample per wave32

// One wave per sample: 32 lanes stream one feature row + the gathered center
// row as contiguous float4s (512B/coalesced load), accumulating diff^2.
// The 32 per-lane partials are reduced with one V_WMMA_F32_16X16X4_F32
// against an all-ones B matrix: A[m][0]=p(lane m), A[m][2]=p(lane m+16),
// so D[m][*] = p(lane m)+p(lane m+16); summing D over rows on lanes 0 and 16
// yields the two 16-lane half-sums, combined in LDS.
__global__ __launch_bounds__(BLOCK_THREADS)
void ProximityLoss_partial_kernel(const float* __restrict__ center,
                                  const float* __restrict__ features,
                                  const int*   __restrict__ labels,
                                  float*       __restrict__ partials)
{
    const int tid   = threadIdx.x;
    const int lane  = tid & 31;
    const int wave  = tid >> 5;
    const int khalf = lane >> 4;
    const int s     = blockIdx.x * SAMPLES_PER_BLOCK + wave;

    const int lab = labels[s];   // uniform per wave -> scalar load

    const v4f* __restrict__ fp = (const v4f*)(features + (size_t)s   * DDIM) + lane;
    const v4f* __restrict__ cp = (const v4f*)(center   + (size_t)lab * DDIM) + lane;

    float acc0 = 0.f, acc1 = 0.f, acc2 = 0.f, acc3 = 0.f;
    #pragma unroll
    for (int j = 0; j < DDIM / 128; ++j) {          // 8 iters, stride 32 float4s
        v4f f = __builtin_nontemporal_load(fp + j * 32);  // features: single-use
        v4f c = cp[j * 32];                               // centers: hot in L2
        float d0 = f.x - c.x;
        float d1 = f.y - c.y;
        float d2 = f.z - c.z;
        float d3 = f.w - c.w;
        acc0 = fmaf(d0, d0, acc0);
        acc1 = fmaf(d1, d1, acc1);
        acc2 = fmaf(d2, d2, acc2);
        acc3 = fmaf(d3, d3, acc3);
    }
    float partial = (acc0 + acc1) + (acc2 + acc3);

    // Cross-lane reduction via WMMA (EXEC all-ones here: uniform control flow).
    v2f a; a.x = partial; a.y = 0.0f;
    v2f b; b.x = 1.0f;    b.y = 1.0f;
    v8f d = {};
    // 8 args: (neg_a, A, neg_b, B, c_mod, C, reuse_a, reuse_b)
    d = __builtin_amdgcn_wmma_f32_16x16x4_f32(false, a, false, b,
                                              (short)0, d, false, false);
    // lane n<16: d[v] = rows 0..7; lane n>=16: rows 8..15 (columns replicated).
    float local = ((d[0] + d[1]) + (d[2] + d[3])) + ((d[4] + d[5]) + (d[6] + d[7]));

    __shared__ float red[WAVES_PER_BLOCK * 2];
    if ((lane & 15) == 0) red[wave * 2 + khalf] = local;  // lanes 0 and 16
    __syncthreads();
    if (tid == 0) {
        float bsum = 0.f;
        #pragma unroll
        for (int w = 0; w < WAVES_PER_BLOCK; ++w) {
            float d2   = red[2 * w] + red[2 * w + 1];     // full ||f-c||^2
            float dist = sqrtf(fmaxf(d2, 0.0f));
            dist = fminf(fmaxf(dist, 1e-12f), 1e12f);
            bsum += dist;
        }
        partials[blockIdx.x] = bsum;
    }
}

// Deterministic single-block tree reduction -> mean.
__global__ __launch_bounds__(128)
void ProximityLoss_reduce_kernel(const float* __restrict__ partials, int n,
                                 float invB, float* __restrict__ out)
{
    __shared__ float smem[128];
    const int t = threadIdx.x;
    float a = 0.f;
    for (int i = t; i < n; i += 128) a += partials[i];   // fixed order per thread
    smem[t] = a;
    __syncthreads();
    #pragma unroll
    for (int off = 64; off > 0; off >>= 1) {
        if (t < off) smem[t] += smem[t + off];
        __syncthreads();
    }
    if (t == 0) out[0] = smem[0] * invB;
}

extern "C" void kernel_launch(void* const* d_in, const int* in_sizes, int n_in,
                              void* d_out, int out_size, void* d_ws, size_t ws_size,
                              hipStream_t stream)
{
    const float* center   = (const float*)d_in[0];  // [C, D] f32
    const float* features = (const float*)d_in[1];  // [B, D] f32
    const int*   labels   = (const int*)  d_in[2];  // [B] int
    float*       out      = (float*)d_out;          // scalar f32

    const int B       = in_sizes[2];                 // 16384
    const int nBlocks = B / SAMPLES_PER_BLOCK;       // 2048

    float* partials = (float*)d_ws;                  // nBlocks floats of scratch

    ProximityLoss_partial_kernel<<<nBlocks, BLOCK_THREADS, 0, stream>>>(
        center, features, labels, partials);
    ProximityLoss_reduce_kernel<<<1, 128, 0, stream>>>(
        partials, nBlocks, 1.0f / (float)B, out);
}